// ShiftAug_11605001634125
// MI455X (gfx1250) — compile-verified
//
#include <hip/hip_runtime.h>

// ShiftAug: out[n,c,y,x] = x[n,c, clip(y+sy-PAD,0,H-1), clip(x+sx-PAD,0,W-1)]
// Bandwidth-bound gather (~260 MB, ~11 us at 23.3 TB/s). Zero FLOPs => no WMMA.
// Strategy per block = one (n,c) plane:
//   1) async-copy the 28 KB plane global->LDS with global_load_async_to_lds_b128
//      (aligned 16B loads, no VGPR round-trip, ASYNCcnt-tracked)
//   2) resolve the replicate-pad shift from LDS (ds_load_b32 gathers)
//   3) non-temporal 128-bit stores to the output stream

typedef __attribute__((ext_vector_type(4))) float v4f;

namespace {
constexpr int kPad        = 3;
constexpr int kN          = 512;
constexpr int kC          = 9;
constexpr int kH          = 84;
constexpr int kW          = 84;
constexpr int kPlane      = kH * kW;        // 7056 floats = 28224 B per plane
constexpr int kVec4PerRow = kW / 4;         // 21
constexpr int kVec4PerPln = kPlane / 4;     // 1764
constexpr int kThreads    = 256;            // 8 wave32 waves
constexpr int kRowStep    = kThreads / kVec4PerRow;  // 12 rows per 256-vec4 step
constexpr int kColStep    = kThreads % kVec4PerRow;  // +4 vec4 per step
}

__global__ __launch_bounds__(kThreads) void ShiftAug_11605001634125_kernel(
    const float* __restrict__ x,
    const int*   __restrict__ shift,
    float*       __restrict__ out)
{
    __shared__ float tile[kPlane];

    const int plane = blockIdx.x;            // plane = n*kC + c  (uniform)
    const int n     = plane / kC;
    const int dx    = shift[2 * n + 0] - kPad;   // column shift (sx) -> s_load
    const int dy    = shift[2 * n + 1] - kPad;   // row shift    (sy) -> s_load

    const float* __restrict__ src = x   + (size_t)plane * kPlane;
    float*       __restrict__ dst = out + (size_t)plane * kPlane;

    // ---- Phase 1: plane -> LDS via async tensor-path copies (ASYNCcnt) ----
    // Low 32 bits of the flat shared pointer == LDS byte offset (local->flat
    // addrspacecast is {aperture_hi32, lds_offset32}).
    const unsigned ldsBase = (unsigned)(uintptr_t)&tile[0];
    for (int q = threadIdx.x; q < kVec4PerPln; q += kThreads) {
        const float* g = src + 4 * q;                 // 16B-aligned
        unsigned     l = ldsBase + (unsigned)(16 * q);
        asm volatile("global_load_async_to_lds_b128 %0, %1, off"
                     :: "v"(l), "v"(g) : "memory");
    }
    asm volatile("s_wait_asynccnt 0x0" ::: "memory");
    __syncthreads();

    // ---- Phase 2: shifted replicate-pad gather from LDS, NT b128 stores ----
    int q  = threadIdx.x;
    int y  = q / kVec4PerRow;          // one division, then incremental update
    int xq = q - y * kVec4PerRow;
    while (q < kVec4PerPln) {
        int ry = y + dy;
        ry = ry < 0 ? 0 : (ry > kH - 1 ? kH - 1 : ry);
        const float* __restrict__ row = tile + ry * kW;

        const int xb = xq * 4 + dx;
        int r0 = xb + 0; r0 = r0 < 0 ? 0 : (r0 > kW - 1 ? kW - 1 : r0);
        int r1 = xb + 1; r1 = r1 < 0 ? 0 : (r1 > kW - 1 ? kW - 1 : r1);
        int r2 = xb + 2; r2 = r2 < 0 ? 0 : (r2 > kW - 1 ? kW - 1 : r2);
        int r3 = xb + 3; r3 = r3 < 0 ? 0 : (r3 > kW - 1 ? kW - 1 : r3);

        v4f v;
        v.x = row[r0];
        v.y = row[r1];
        v.z = row[r2];
        v.w = row[r3];

        // 16B-aligned (row stride 336B = 21*16B) -> global_store_b128 NT.
        __builtin_nontemporal_store(v, reinterpret_cast<v4f*>(dst + y * kW + xq * 4));

        q += kThreads;
        y += kRowStep;
        xq += kColStep;
        if (xq >= kVec4PerRow) { xq -= kVec4PerRow; ++y; }
    }
}

extern "C" void kernel_launch(void* const* d_in, const int* in_sizes, int n_in,
                              void* d_out, int out_size, void* d_ws, size_t ws_size,
                              hipStream_t stream) {
    (void)in_sizes; (void)n_in; (void)out_size; (void)d_ws; (void)ws_size;
    const float* x     = (const float*)d_in[0];
    const int*   shift = (const int*)d_in[1];
    float*       out   = (float*)d_out;

    dim3 grid(kN * kC);     // 4608 planes
    dim3 block(kThreads);
    hipLaunchKernelGGL(ShiftAug_11605001634125_kernel, grid, block, 0, stream,
                       x, shift, out);
}